// GNN_individual_11647951306806
// MI455X (gfx1250) — compile-verified
//
#include <hip/hip_runtime.h>

typedef __attribute__((ext_vector_type(2))) float v2f;
typedef __attribute__((ext_vector_type(8))) float v8f;

#define NF 64   // input features
#define NH 64   // output features

// ---------------------------------------------------------------------------
// Kernel 1: out[n,o] = bias[o]   ;   deg[n] = 1 (self-loop)
// ---------------------------------------------------------------------------
__global__ void gcn_init_kernel(float* __restrict__ out,
                                const float* __restrict__ bias,
                                unsigned int* __restrict__ deg,
                                int n) {
    int stride = gridDim.x * blockDim.x;
    int total  = n * NH;
    for (int t = blockIdx.x * blockDim.x + threadIdx.x; t < total; t += stride) {
        out[t] = bias[t & (NH - 1)];
        if (t < n) deg[t] = 1u;
    }
}

// ---------------------------------------------------------------------------
// Kernel 2: deg[dst[e]] += 1
// ---------------------------------------------------------------------------
__global__ void gcn_degree_kernel(const int* __restrict__ dst,
                                  unsigned int* __restrict__ deg,
                                  int E) {
    int e = blockIdx.x * blockDim.x + threadIdx.x;
    if (e < E) atomicAdd(&deg[dst[e]], 1u);
}

// ---------------------------------------------------------------------------
// Kernel 3: dinv[i] = rsqrt(deg[i])   (deg >= 1 always: self-loop included)
// ---------------------------------------------------------------------------
__global__ void gcn_dinv_kernel(const unsigned int* __restrict__ deg,
                                float* __restrict__ dinv,
                                int n) {
    int i = blockIdx.x * blockDim.x + threadIdx.x;
    if (i < n) dinv[i] = rsqrtf((float)deg[i]);
}

// ---------------------------------------------------------------------------
// Kernel 4: h = x @ W^T  via V_WMMA_F32_16X16X4_F32 (wave32, one wave = 16x64)
//
// A (16x4 f32) layout : lanes 0-15 -> M=lane, v0=K0,v1=K1 ; lanes 16-31 -> K2,K3
// B (4x16  f32) layout: lanes 0-15 -> N=lane, v0=K0,v1=K1 ; lanes 16-31 -> K2,K3
//       B[k][n] = W^T[k][n] = W[n][k]  (W row-major [NH, NF])
// C/D (16x16 f32): VGPR j: lanes 0-15 -> M=j, N=lane ; lanes 16-31 -> M=j+8
// ---------------------------------------------------------------------------
__global__ void gcn_gemm_wmma_kernel(const float* __restrict__ x,
                                     const float* __restrict__ w,
                                     float* __restrict__ h,
                                     int ntiles) {
    __shared__ float wl[NH * NF];   // 16 KB: full weight matrix in LDS
    for (int t = threadIdx.x; t < NH * NF; t += blockDim.x)
        wl[t] = w[t];
    __syncthreads();

    const int wave = (blockIdx.x * blockDim.x + threadIdx.x) >> 5;
    if (wave >= ntiles) return;                 // wave-uniform: EXEC stays all-1s

    const int lane    = threadIdx.x & 31;
    const int lid     = lane & 15;              // M (A) / N (B) index
    const int halfSel = lane >> 4;              // 0: K pair {0,1}, 1: K pair {2,3}
    const int m0      = wave * 16;

    const v8f vzero = {};
    v8f acc[4];
#pragma unroll
    for (int nt = 0; nt < 4; ++nt) acc[nt] = vzero;

    const float* xrow = x + (size_t)(m0 + lid) * NF + 2 * halfSel;

    for (int k0 = 0; k0 < NF; k0 += 4) {
        v2f a = *(const v2f*)(xrow + k0);       // A fragment (8B aligned)
#pragma unroll
        for (int nt = 0; nt < 4; ++nt) {
            int col = nt * 16 + lid;
            v2f b = *(const v2f*)&wl[col * NF + k0 + 2 * halfSel];
            acc[nt] = __builtin_amdgcn_wmma_f32_16x16x4_f32(
                /*neg_a=*/false, a, /*neg_b=*/false, b,
                /*c_mod=*/(short)0, acc[nt],
                /*reuse_a=*/false, /*reuse_b=*/false);
        }
    }

    // store D: row = m0 + j + 8*halfSel, col = nt*16 + lid
#pragma unroll
    for (int nt = 0; nt < 4; ++nt) {
        int col = nt * 16 + lid;
#pragma unroll
        for (int j = 0; j < 8; ++j) {
            int row = m0 + j + 8 * halfSel;
            h[(size_t)row * NH + col] = acc[nt][j];
        }
    }
}

// Scalar tail for rows not covered by full 16-row tiles (n % 16 rows; 0 here).
__global__ void gcn_gemm_tail_kernel(const float* __restrict__ x,
                                     const float* __restrict__ w,
                                     float* __restrict__ h,
                                     int row0, int n) {
    int t = blockIdx.x * blockDim.x + threadIdx.x;
    int row = row0 + (t >> 6);
    int o   = t & 63;
    if (row >= n) return;
    float s = 0.f;
    for (int k = 0; k < NF; ++k)
        s += x[(size_t)row * NF + k] * w[o * NF + k];
    h[(size_t)row * NH + o] = s;
}

// ---------------------------------------------------------------------------
// Kernel 5: scatter-add.  One wave32 per (edge | self-loop); each lane owns a
// float2 slice of the 64-wide feature row (coalesced 256B), 2 atomics/lane.
// Entries e < E are graph edges; e in [E, E+n) are self-loops (s = d = e - E).
// ---------------------------------------------------------------------------
__global__ void gcn_scatter_kernel(const int* __restrict__ src,
                                   const int* __restrict__ dst,
                                   const float* __restrict__ h,
                                   const float* __restrict__ dinv,
                                   float* __restrict__ out,
                                   int E, int n) {
    long long gt = (long long)blockIdx.x * blockDim.x + threadIdx.x;
    int e = (int)(gt >> 5);
    if (e >= E + n) return;
    int lane = threadIdx.x & 31;

    int s, d;
    if (e < E) { s = src[e]; d = dst[e]; }
    else       { s = d = e - E; }

    float nrm = dinv[s] * dinv[d];
    v2f hv = *(const v2f*)&h[(size_t)s * NH + lane * 2];
    float* o = &out[(size_t)d * NH + lane * 2];
    atomicAdd(o + 0, hv.x * nrm);
    atomicAdd(o + 1, hv.y * nrm);
}

// ---------------------------------------------------------------------------
extern "C" void kernel_launch(void* const* d_in, const int* in_sizes, int n_in,
                              void* d_out, int out_size, void* d_ws, size_t ws_size,
                              hipStream_t stream) {
    const float* x    = (const float*)d_in[0];
    const int*   edge = (const int*)d_in[1];   // [2, E] int
    const float* w    = (const float*)d_in[2]; // [NH, NF]
    const float* bias = (const float*)d_in[3]; // [NH]
    float*       out  = (float*)d_out;

    const int n = in_sizes[0] / NF;
    const int E = in_sizes[1] / 2;
    const int* srcIdx = edge;
    const int* dstIdx = edge + E;

    // workspace: deg (u32[n]) | dinv (f32[n]) | h (f32[n*NH]), 256B aligned
    size_t nb      = (size_t)n;
    size_t dinvOff = (nb * 4 + 255) & ~(size_t)255;
    size_t hOff    = (dinvOff + nb * 4 + 255) & ~(size_t)255;
    unsigned int* deg  = (unsigned int*)d_ws;
    float*        dinv = (float*)((char*)d_ws + dinvOff);
    float*        h    = (float*)((char*)d_ws + hOff);

    const int BLK = 256;

    // 1) out = bias, deg = 1
    {
        int total = n * NH;
        int grid  = (total + BLK - 1) / BLK;
        gcn_init_kernel<<<grid, BLK, 0, stream>>>(out, bias, deg, n);
    }
    // 2) in-degree
    gcn_degree_kernel<<<(E + BLK - 1) / BLK, BLK, 0, stream>>>(dstIdx, deg, E);
    // 3) dinv
    gcn_dinv_kernel<<<(n + BLK - 1) / BLK, BLK, 0, stream>>>(deg, dinv, n);
    // 4) h = x @ W^T (WMMA full tiles + scalar tail)
    {
        int ntiles = n / 16;                       // full 16-row tiles
        if (ntiles > 0) {
            int wavesPerBlk = BLK / 32;
            int grid = (ntiles + wavesPerBlk - 1) / wavesPerBlk;
            gcn_gemm_wmma_kernel<<<grid, BLK, 0, stream>>>(x, w, h, ntiles);
        }
        int rem = n - ntiles * 16;
        if (rem > 0) {
            int total = rem * NH;
            gcn_gemm_tail_kernel<<<(total + BLK - 1) / BLK, BLK, 0, stream>>>(
                x, w, h, ntiles * 16, n);
        }
    }
    // 5) scatter (edges + self-loops)
    {
        long long total = (long long)(E + n) * 32;
        long long grid  = (total + BLK - 1) / BLK;
        gcn_scatter_kernel<<<(int)grid, BLK, 0, stream>>>(srcIdx, dstIdx, h,
                                                          dinv, out, E, n);
    }
}